// GCNLayer_concatenate_1486058684814
// MI455X (gfx1250) — compile-verified
//
#include <hip/hip_runtime.h>

typedef __attribute__((ext_vector_type(2))) float v2f;
typedef __attribute__((ext_vector_type(8))) float v8f;

#define DIM 64          // feature width / output width
#define K2  128         // concat width
#define HSTRIDE 130     // LDS row stride (floats): odd*2 -> conflict-free b64 reads

// ---------------- zero the aggregation buffer ----------------
__global__ __launch_bounds__(256) void zero_kernel(float4* __restrict__ p, int n4) {
  int i = blockIdx.x * 256 + threadIdx.x;
  if (i < n4) p[i] = make_float4(0.f, 0.f, 0.f, 0.f);
}

// ---------------- edge scatter-add: agg[dst] += feature[src] ----------------
// 16 threads per edge, one float4 each -> fully coalesced gather, f32 atomics
// into an L2-resident 25.6MB buffer.
__global__ __launch_bounds__(256) void scatter_kernel(const float* __restrict__ feature,
                                                      const int* __restrict__ src,
                                                      const int* __restrict__ dst,
                                                      float* __restrict__ agg,
                                                      int n_edges) {
  int i = blockIdx.x * 256 + threadIdx.x;
  int total = n_edges * 16;
  if (i >= total) return;
  int e = i >> 4;
  int c = i & 15;                       // which float4 of the 64-wide row
  int s = src[e];
  int d = dst[e];
  float4 v = reinterpret_cast<const float4*>(feature)[(size_t)s * 16 + c];
  float* p = agg + (size_t)d * DIM + c * 4;
  atomicAdd(p + 0, v.x);
  atomicAdd(p + 1, v.y);
  atomicAdd(p + 2, v.z);
  atomicAdd(p + 3, v.w);
}

// ---------------- fused concat + GEMM + bias, fp32 WMMA ----------------
// out[n, o] = sum_k h[n,k] * W[o,k] + b[o],  h = [feature | agg]
// Block: 128 threads = 4 waves, one 16-node tile. Wave w computes output
// columns [16w, 16w+16) via 32 x V_WMMA_F32_16X16X4_F32 over K=128.
__global__ __launch_bounds__(128) void gemm_wmma_kernel(const float* __restrict__ feature,
                                                        const float* __restrict__ agg,
                                                        const float* __restrict__ W,     // [64,128] row-major
                                                        const float* __restrict__ bias,  // [64]
                                                        float* __restrict__ out,         // [n,64]
                                                        int n_nodes) {
  __shared__ float hT[16 * HSTRIDE];    // 16 x 128 concat tile, padded
  const int tid   = threadIdx.x;
  const int node0 = blockIdx.x * 16;

  // Cooperative stage of h-tile: each thread loads a 16-float segment (4 x float4).
  {
    int r  = tid >> 3;                  // row 0..15
    int cb = (tid & 7) * 16;            // col base 0..112
    int node = node0 + r;
    if (node >= n_nodes) node = n_nodes - 1;   // clamp (tail safety)
    const float* srcp = (cb < DIM)
        ? (feature + (size_t)node * DIM + cb)
        : (agg     + (size_t)node * DIM + (cb - DIM));
    float* dstp = hT + r * HSTRIDE + cb;
#pragma unroll
    for (int q = 0; q < 4; ++q) {
      float4 v = reinterpret_cast<const float4*>(srcp)[q];
      dstp[4 * q + 0] = v.x;
      dstp[4 * q + 1] = v.y;
      dstp[4 * q + 2] = v.z;
      dstp[4 * q + 3] = v.w;
    }
  }
  __syncthreads();

  const int wave = tid >> 5;            // 0..3 -> output col tile
  const int lane = tid & 31;
  const int half = lane >> 4;           // 0: lanes 0-15, 1: lanes 16-31
  const int l16  = lane & 15;

  v8f acc = {};
  // A fragment row for this lane (M = l16); B row of W for this lane (N = l16).
  const float* arow = hT + l16 * HSTRIDE;
  const float* wrow = W + (size_t)(wave * 16 + l16) * K2;

#pragma unroll
  for (int kk = 0; kk < 32; ++kk) {
    const int k0 = kk * 4;
    // fp32 16x4 A layout: VGPR0 = K {k0 | k0+2}, VGPR1 = K {k0+1 | k0+3}
    v2f a, b;
    a.x = arow[k0 + 2 * half + 0];
    a.y = arow[k0 + 2 * half + 1];
    // B 4x16 row-striped: same K split across lane halves, N = lane
    b.x = wrow[k0 + 2 * half + 0];
    b.y = wrow[k0 + 2 * half + 1];
    acc = __builtin_amdgcn_wmma_f32_16x16x4_f32(false, a, false, b,
                                                (short)0, acc, false, false);
  }

  // C/D layout: VGPR j -> M = j + 8*half, N = l16
  const int o = wave * 16 + l16;
  const float bo = bias[o];
#pragma unroll
  for (int j = 0; j < 8; ++j) {
    int node = node0 + j + 8 * half;
    if (node < n_nodes)
      out[(size_t)node * DIM + o] = acc[j] + bo;
  }
}

extern "C" void kernel_launch(void* const* d_in, const int* in_sizes, int n_in,
                              void* d_out, int out_size, void* d_ws, size_t ws_size,
                              hipStream_t stream) {
  const float* feature = (const float*)d_in[0];
  const int*   src     = (const int*)d_in[1];
  const int*   dst     = (const int*)d_in[2];
  const float* W       = (const float*)d_in[3];
  const float* bias    = (const float*)d_in[4];
  float*       out     = (float*)d_out;

  const int n_nodes = in_sizes[0] / DIM;
  const int n_edges = in_sizes[1];
  float* agg = (float*)d_ws;            // n_nodes * 64 fp32 = 25.6 MB scratch

  const int n4 = n_nodes * (DIM / 4);
  zero_kernel<<<(n4 + 255) / 256, 256, 0, stream>>>((float4*)agg, n4);

  const long long tscat = (long long)n_edges * 16;
  const int sblocks = (int)((tscat + 255) / 256);
  scatter_kernel<<<sblocks, 256, 0, stream>>>(feature, src, dst, agg, n_edges);

  const int ntiles = (n_nodes + 15) / 16;
  gemm_wmma_kernel<<<ntiles, 128, 0, stream>>>(feature, agg, W, bias, out, n_nodes);
}